// DiversificationGAT_15530601742631
// MI455X (gfx1250) — compile-verified
//
#include <hip/hip_runtime.h>
#include <math.h>

#define N_NODES 50000
#define N_EDGES 800000
#define E_TOT   (N_EDGES + N_NODES)   // 850000 after self-loops
#define IN_DIM  128
#define F1      256                    // HEADS*HID
#define HEADS   4
#define NEG_SLOPE 0.2f

typedef float v2f __attribute__((ext_vector_type(2)));
typedef float v4f __attribute__((ext_vector_type(4)));
typedef float v8f __attribute__((ext_vector_type(8)));

// ---------------- float atomic max via monotone int mapping ----------------
__device__ __forceinline__ void atomicMaxFloat(float* addr, float val) {
    if (val >= 0.0f) {
        atomicMax((int*)addr, __float_as_int(val));
    } else {
        atomicMin((unsigned int*)addr, (unsigned int)__float_as_int(val));
    }
}

// ---------------- init: fill raw bit pattern (used for -inf) ----------------
__global__ void fill_bits_kernel(unsigned int* __restrict__ p, unsigned int v, int n) {
    int i = blockIdx.x * blockDim.x + threadIdx.x;
    if (i < n) p[i] = v;
}

// ---------------- fp32 GEMM via V_WMMA_F32_16X16X4_F32 ----------------
// C[M,256] = A[M,128] @ B[128,256]; one wave computes one 16x16 tile.
__global__ __launch_bounds__(128) void gemm128_wmma(
    const float* __restrict__ A, const float* __restrict__ B,
    float* __restrict__ C, int M)
{
    const int lane = threadIdx.x & 31;
    const int wave = blockIdx.x * (blockDim.x >> 5) + (threadIdx.x >> 5);
    const int tn = wave & 15;            // 256/16 = 16 N tiles
    const int tm = wave >> 4;
    if (tm * 16 >= M) return;

    const int r  = lane & 15;            // row within tile (A) / col within tile (B)
    const int kh = (lane >> 4) << 1;     // lanes 16-31 hold K+2,K+3 halves

    const float* ap = A + (size_t)(tm * 16 + r) * IN_DIM + kh;
    const float* bp = B + (size_t)kh * F1 + tn * 16 + r;

    v8f acc = {};
    #pragma unroll
    for (int k = 0; k < IN_DIM; k += 4) {
        v2f a = *(const v2f*)(ap + k);   // A[r][k+kh], A[r][k+kh+1]
        v2f b;
        b.x = bp[k * F1];                // B[k+kh][n]
        b.y = bp[k * F1 + F1];           // B[k+kh+1][n]
        acc = __builtin_amdgcn_wmma_f32_16x16x4_f32(
            false, a, false, b, (short)0, acc, false, false);
    }
    // C/D layout: VGPR v -> M = v (lanes 0-15) or v+8 (lanes 16-31), N = lane&15
    const int n  = tn * 16 + r;
    const int mb = tm * 16 + ((lane >> 4) << 3);
    #pragma unroll
    for (int v = 0; v < 8; ++v)
        C[(size_t)(mb + v) * F1 + n] = acc[v];
}

// ---------------- mean of edge_attr over original edges ----------------
__global__ __launch_bounds__(256) void edge_attr_mean(
    const float* __restrict__ ea, float* __restrict__ eas)
{
    float s0 = 0.f, s1 = 0.f, s2 = 0.f;
    for (int e = blockIdx.x * blockDim.x + threadIdx.x; e < N_EDGES;
         e += gridDim.x * blockDim.x) {
        s0 += ea[e * 3 + 0]; s1 += ea[e * 3 + 1]; s2 += ea[e * 3 + 2];
    }
    for (int m = 16; m; m >>= 1) {
        s0 += __shfl_xor(s0, m, 32);
        s1 += __shfl_xor(s1, m, 32);
        s2 += __shfl_xor(s2, m, 32);
    }
    __shared__ float sm[3][8];
    const int w = threadIdx.x >> 5, l = threadIdx.x & 31;
    if (l == 0) { sm[0][w] = s0; sm[1][w] = s1; sm[2][w] = s2; }
    __syncthreads();
    if (threadIdx.x == 0) {
        float t0 = 0.f, t1 = 0.f, t2 = 0.f;
        for (int i = 0; i < 8; ++i) { t0 += sm[0][i]; t1 += sm[1][i]; t2 += sm[2][i]; }
        atomicAdd(&eas[0], t0); atomicAdd(&eas[1], t1); atomicAdd(&eas[2], t2);
    }
}

// ---------------- layer-1 edge pass A: logits + segment max ----------------
// One wave per edge; lane owns 8 contiguous channels (head = lane>>3).
__global__ __launch_bounds__(256) void edge1_logits(
    const int* __restrict__ ei, const float* __restrict__ ea,
    const float* __restrict__ xl, const float* __restrict__ xr,
    const float* __restrict__ W1e, const float* __restrict__ att1,
    const float* __restrict__ eas,
    float* __restrict__ log1, float* __restrict__ mx1)
{
    __shared__ float s_att[F1];
    __shared__ float s_we[3 * F1];
    for (int i = threadIdx.x; i < F1; i += blockDim.x) s_att[i] = att1[i];
    for (int i = threadIdx.x; i < 3 * F1; i += blockDim.x) s_we[i] = W1e[i];
    __syncthreads();

    const int lane = threadIdx.x & 31;
    const int e = blockIdx.x * 8 + (threadIdx.x >> 5);
    if (e >= E_TOT) return;

    int src, dst; float a0, a1, a2;
    if (e < N_EDGES) {
        src = ei[e]; dst = ei[N_EDGES + e];
        a0 = ea[e * 3]; a1 = ea[e * 3 + 1]; a2 = ea[e * 3 + 2];
    } else {
        src = dst = e - N_EDGES;
        const float inv = 1.0f / (float)N_EDGES;
        a0 = eas[0] * inv; a1 = eas[1] * inv; a2 = eas[2] * inv;
    }

    const int c0 = lane * 8;
    v4f xla = *(const v4f*)(xl + (size_t)src * F1 + c0);
    v4f xlb = *(const v4f*)(xl + (size_t)src * F1 + c0 + 4);
    v4f xra = *(const v4f*)(xr + (size_t)dst * F1 + c0);
    v4f xrb = *(const v4f*)(xr + (size_t)dst * F1 + c0 + 4);

    float p = 0.f;
    #pragma unroll
    for (int j = 0; j < 8; ++j) {
        const int c = c0 + j;
        const float xs = (j < 4 ? xla[j] : xlb[j - 4]) + (j < 4 ? xra[j] : xrb[j - 4]);
        const float ef = a0 * s_we[c] + a1 * s_we[F1 + c] + a2 * s_we[2 * F1 + c];
        float m = xs + ef;
        m = m > 0.f ? m : NEG_SLOPE * m;
        p += s_att[c] * m;
    }
    p += __shfl_xor(p, 1, 32);
    p += __shfl_xor(p, 2, 32);
    p += __shfl_xor(p, 4, 32);
    if ((lane & 7) == 0) {
        const int h = lane >> 3;
        log1[(size_t)e * HEADS + h] = p;
        atomicMaxFloat(&mx1[dst * HEADS + h], p);
    }
}

// ---------------- layer-1 edge pass B: exp + segment sum ----------------
__global__ void edge1_exp(
    const int* __restrict__ ei, float* __restrict__ log1,
    const float* __restrict__ mx1, float* __restrict__ den1)
{
    const int idx = blockIdx.x * blockDim.x + threadIdx.x;
    if (idx >= E_TOT * HEADS) return;
    const int e = idx >> 2, h = idx & 3;
    const int dst = (e < N_EDGES) ? ei[N_EDGES + e] : (e - N_EDGES);
    const float a = expf(log1[idx] - mx1[dst * HEADS + h]);
    log1[idx] = a;                         // overwrite logits with exp values
    atomicAdd(&den1[dst * HEADS + h], a);
}

// ---------------- layer-1 edge pass C: weighted aggregation ----------------
__global__ __launch_bounds__(256) void edge1_aggregate(
    const int* __restrict__ ei, const float* __restrict__ xl,
    const float* __restrict__ a1, const float* __restrict__ den1,
    float* __restrict__ out1)
{
    const int lane = threadIdx.x & 31;
    const int e = blockIdx.x * 8 + (threadIdx.x >> 5);
    if (e >= E_TOT) return;
    int src, dst;
    if (e < N_EDGES) { src = ei[e]; dst = ei[N_EDGES + e]; }
    else             { src = dst = e - N_EDGES; }
    const int h = lane >> 3;
    const float alpha = a1[(size_t)e * HEADS + h] / den1[dst * HEADS + h];
    const int c0 = lane * 8;
    const float* xs = xl + (size_t)src * F1 + c0;
    float* od = out1 + (size_t)dst * F1 + c0;
    #pragma unroll
    for (int j = 0; j < 8; ++j)
        atomicAdd(&od[j], xs[j] * alpha);
}

// ---------------- node: ELU + fused 256->1 projections for layer 2 ----------
__global__ __launch_bounds__(256) void node_elu_lin2(
    const float* __restrict__ out1, const float* __restrict__ b1,
    const float* __restrict__ W2l, const float* __restrict__ W2r,
    float* __restrict__ sl, float* __restrict__ sr)
{
    const int lane = threadIdx.x & 31;
    const int n = blockIdx.x * 8 + (threadIdx.x >> 5);
    if (n >= N_NODES) return;
    const int c0 = lane * 8;
    float pl = 0.f, pr = 0.f;
    #pragma unroll
    for (int j = 0; j < 8; ++j) {
        const int c = c0 + j;
        float v = out1[(size_t)n * F1 + c] + b1[c];
        v = v > 0.f ? v : (expf(v) - 1.f);     // ELU
        pl += v * W2l[c];
        pr += v * W2r[c];
    }
    for (int m = 1; m < 32; m <<= 1) {
        pl += __shfl_xor(pl, m, 32);
        pr += __shfl_xor(pr, m, 32);
    }
    if (lane == 0) { sl[n] = pl; sr[n] = pr; }
}

// ---------------- layer-2 edge passes (scalar per edge) ----------------
__global__ void edge2_logits(
    const int* __restrict__ ei, const float* __restrict__ ea,
    const float* __restrict__ W2e, const float* __restrict__ att2,
    const float* __restrict__ eas,
    const float* __restrict__ sl, const float* __restrict__ sr,
    float* __restrict__ log2, float* __restrict__ mx2)
{
    const int e = blockIdx.x * blockDim.x + threadIdx.x;
    if (e >= E_TOT) return;
    int src, dst; float a0, a1, a2;
    if (e < N_EDGES) {
        src = ei[e]; dst = ei[N_EDGES + e];
        a0 = ea[e * 3]; a1 = ea[e * 3 + 1]; a2 = ea[e * 3 + 2];
    } else {
        src = dst = e - N_EDGES;
        const float inv = 1.0f / (float)N_EDGES;
        a0 = eas[0] * inv; a1 = eas[1] * inv; a2 = eas[2] * inv;
    }
    const float ef = a0 * W2e[0] + a1 * W2e[1] + a2 * W2e[2];
    float m = sl[src] + sr[dst] + ef;
    m = m > 0.f ? m : NEG_SLOPE * m;
    const float lg = att2[0] * m;
    log2[e] = lg;
    atomicMaxFloat(&mx2[dst], lg);
}

__global__ void edge2_exp(
    const int* __restrict__ ei, float* __restrict__ log2,
    const float* __restrict__ mx2, float* __restrict__ den2)
{
    const int e = blockIdx.x * blockDim.x + threadIdx.x;
    if (e >= E_TOT) return;
    const int dst = (e < N_EDGES) ? ei[N_EDGES + e] : (e - N_EDGES);
    const float a = expf(log2[e] - mx2[dst]);
    log2[e] = a;
    atomicAdd(&den2[dst], a);
}

__global__ void edge2_aggregate(
    const int* __restrict__ ei, const float* __restrict__ sl,
    const float* __restrict__ a2, const float* __restrict__ den2,
    float* __restrict__ out2)
{
    const int e = blockIdx.x * blockDim.x + threadIdx.x;
    if (e >= E_TOT) return;
    int src, dst;
    if (e < N_EDGES) { src = ei[e]; dst = ei[N_EDGES + e]; }
    else             { src = dst = e - N_EDGES; }
    atomicAdd(&out2[dst], sl[src] * a2[e] / den2[dst]);
}

// ---------------- global softmax over gat_scores ----------------
__global__ __launch_bounds__(256) void scores_max(
    const float* __restrict__ out2, const float* __restrict__ b2,
    float* __restrict__ dout, float* __restrict__ mx3)
{
    const int n = blockIdx.x * blockDim.x + threadIdx.x;
    float s = -INFINITY;
    if (n < N_NODES) { s = out2[n] + b2[0]; dout[N_NODES + n] = s; }
    for (int m = 16; m; m >>= 1) s = fmaxf(s, __shfl_xor(s, m, 32));
    __shared__ float sm[8];
    if ((threadIdx.x & 31) == 0) sm[threadIdx.x >> 5] = s;
    __syncthreads();
    if (threadIdx.x == 0) {
        float t = sm[0];
        for (int i = 1; i < 8; ++i) t = fmaxf(t, sm[i]);
        atomicMaxFloat(mx3, t);
    }
}

__global__ __launch_bounds__(256) void scores_exp(
    float* __restrict__ dout, const float* __restrict__ mx3,
    float* __restrict__ sum3)
{
    const int n = blockIdx.x * blockDim.x + threadIdx.x;
    float p = 0.f;
    if (n < N_NODES) { p = expf(dout[N_NODES + n] - mx3[0]); dout[n] = p; }
    for (int m = 16; m; m >>= 1) p += __shfl_xor(p, m, 32);
    __shared__ float sm[8];
    if ((threadIdx.x & 31) == 0) sm[threadIdx.x >> 5] = p;
    __syncthreads();
    if (threadIdx.x == 0) {
        float t = 0.f;
        for (int i = 0; i < 8; ++i) t += sm[i];
        atomicAdd(sum3, t);
    }
}

__global__ void scores_norm(float* __restrict__ dout, const float* __restrict__ sum3)
{
    const int n = blockIdx.x * blockDim.x + threadIdx.x;
    if (n < N_NODES) dout[n] /= sum3[0];
}

// ---------------- launch ----------------
extern "C" void kernel_launch(void* const* d_in, const int* in_sizes, int n_in,
                              void* d_out, int out_size, void* d_ws, size_t ws_size,
                              hipStream_t stream)
{
    (void)in_sizes; (void)n_in; (void)out_size; (void)ws_size;

    const float* x    = (const float*)d_in[0];
    const float* ea   = (const float*)d_in[1];
    const float* W1l  = (const float*)d_in[2];
    const float* W1r  = (const float*)d_in[3];
    const float* W1e  = (const float*)d_in[4];
    const float* att1 = (const float*)d_in[5];
    const float* b1   = (const float*)d_in[6];
    const float* W2l  = (const float*)d_in[7];
    const float* W2r  = (const float*)d_in[8];
    const float* W2e  = (const float*)d_in[9];
    const float* att2 = (const float*)d_in[10];
    const float* b2   = (const float*)d_in[11];
    const int*   ei   = (const int*)d_in[12];
    float* out = (float*)d_out;

    // workspace layout (floats)
    float* ws = (float*)d_ws;
    size_t o = 0;
    float* XL1  = ws + o; o += (size_t)N_NODES * F1;       // x @ W1l
    float* XR1  = ws + o; o += (size_t)N_NODES * F1;       // x @ W1r
    float* OUT1 = ws + o; o += (size_t)N_NODES * F1;       // layer-1 aggregation
    float* LOG1 = ws + o; o += (size_t)E_TOT * HEADS;      // logits -> exp vals
    float* LOG2 = ws + o; o += (size_t)E_TOT;
    float* SL   = ws + o; o += N_NODES;
    float* SR   = ws + o; o += N_NODES;
    float* ZERO0= ws + o;                                  // zero-init region start
    float* DEN1 = ws + o; o += (size_t)N_NODES * HEADS;
    float* DEN2 = ws + o; o += N_NODES;
    float* OUT2 = ws + o; o += N_NODES;
    float* EAS  = ws + o; o += 3;
    float* SUM3 = ws + o; o += 1;
    const size_t zeroCount = (size_t)((ws + o) - ZERO0);
    float* MX1  = ws + o; o += (size_t)N_NODES * HEADS;    // -inf init region
    float* MX2  = ws + o; o += N_NODES;
    float* MX3  = ws + o; o += 1;
    const int negN = N_NODES * HEADS + N_NODES + 1;

    // per-call re-init (graph-capture safe)
    hipMemsetAsync(OUT1, 0, (size_t)N_NODES * F1 * sizeof(float), stream);
    hipMemsetAsync(ZERO0, 0, zeroCount * sizeof(float), stream);
    fill_bits_kernel<<<(negN + 255) / 256, 256, 0, stream>>>(
        (unsigned int*)MX1, 0xFF800000u, negN);

    // dense transforms via WMMA (50000/16 * 256/16 = 50000 waves, 4 waves/block)
    gemm128_wmma<<<12500, 128, 0, stream>>>(x, W1l, XL1, N_NODES);
    gemm128_wmma<<<12500, 128, 0, stream>>>(x, W1r, XR1, N_NODES);

    edge_attr_mean<<<512, 256, 0, stream>>>(ea, EAS);

    // layer 1
    edge1_logits<<<(E_TOT + 7) / 8, 256, 0, stream>>>(
        ei, ea, XL1, XR1, W1e, att1, EAS, LOG1, MX1);
    edge1_exp<<<(E_TOT * HEADS + 255) / 256, 256, 0, stream>>>(ei, LOG1, MX1, DEN1);
    edge1_aggregate<<<(E_TOT + 7) / 8, 256, 0, stream>>>(ei, XL1, LOG1, DEN1, OUT1);

    node_elu_lin2<<<(N_NODES + 7) / 8, 256, 0, stream>>>(OUT1, b1, W2l, W2r, SL, SR);

    // layer 2
    edge2_logits<<<(E_TOT + 255) / 256, 256, 0, stream>>>(
        ei, ea, W2e, att2, EAS, SL, SR, LOG2, MX2);
    edge2_exp<<<(E_TOT + 255) / 256, 256, 0, stream>>>(ei, LOG2, MX2, DEN2);
    edge2_aggregate<<<(E_TOT + 255) / 256, 256, 0, stream>>>(ei, SL, LOG2, DEN2, OUT2);

    // global softmax -> portfolio weights
    scores_max<<<(N_NODES + 255) / 256, 256, 0, stream>>>(OUT2, b2, out, MX3);
    scores_exp<<<(N_NODES + 255) / 256, 256, 0, stream>>>(out, MX3, SUM3);
    scores_norm<<<(N_NODES + 255) / 256, 256, 0, stream>>>(out, SUM3);
}